// GraphConvolution_58634893525267
// MI455X (gfx1250) — compile-verified
//
#include <hip/hip_runtime.h>

typedef float v2f __attribute__((ext_vector_type(2)));
typedef float v4f __attribute__((ext_vector_type(4)));
typedef float v8f __attribute__((ext_vector_type(8)));

#define N_NODES 100000
#define N_EDGES 1600000
#define IN_DIM  128
#define OUT_DIM 128
#define ROW_TILES (N_NODES / 16)   // 6250, exact

// ---------------------------------------------------------------------------
// Phase 0: out[n, c] = bias[c]   (atomics accumulate on top of this)
// ---------------------------------------------------------------------------
__global__ __launch_bounds__(256) void gcn_init_out(const float* __restrict__ bias,
                                                    float* __restrict__ out) {
    int i = blockIdx.x * 256 + threadIdx.x;          // float4 index
    v4f b = ((const v4f*)bias)[i & (OUT_DIM / 4 - 1)];
    ((v4f*)out)[i] = b;
}

// ---------------------------------------------------------------------------
// Phase 1: support = X @ W via V_WMMA_F32_16X16X4_F32
//   block = 256 threads = 8 waves; wave w owns column tile n0 = 16*w
//   W staged in LDS (64KB); grid-stride over 16-row tiles of X
// ---------------------------------------------------------------------------
__global__ __launch_bounds__(256) void gcn_gemm(const float* __restrict__ X,
                                                const float* __restrict__ W,
                                                float* __restrict__ S) {
    __shared__ float wlds[IN_DIM * OUT_DIM];         // 64 KB
    for (int i = threadIdx.x; i < IN_DIM * OUT_DIM / 4; i += 256)
        ((v4f*)wlds)[i] = ((const v4f*)W)[i];
    __syncthreads();

    const int lane    = threadIdx.x & 31;
    const int wave    = threadIdx.x >> 5;
    const int n0      = wave * 16;
    const int halfsel = lane >> 4;                   // 0: K=0,1  1: K=2,3
    const int l15     = lane & 15;

    for (int rt = blockIdx.x; rt < ROW_TILES; rt += gridDim.x) {
        const int m0 = rt * 16;
        // A: lane holds row (m0 + l15), two consecutive K values
        const float* arow = X + (size_t)(m0 + l15) * IN_DIM + halfsel * 2;

        v8f acc = {};
        #pragma unroll
        for (int k0 = 0; k0 < IN_DIM; k0 += 4) {
            v2f a = *(const v2f*)(arow + k0);
            v2f b;
            b.x = wlds[(k0 + halfsel * 2 + 0) * OUT_DIM + n0 + l15];
            b.y = wlds[(k0 + halfsel * 2 + 1) * OUT_DIM + n0 + l15];
            // 8 args: (neg_a, A, neg_b, B, c_mod, C, reuse_a, reuse_b)
            acc = __builtin_amdgcn_wmma_f32_16x16x4_f32(
                false, a, false, b, (short)0, acc, false, false);
        }

        // D layout: VGPR r -> M = r + 8*halfsel, N = l15
        float* srow = S + (size_t)(m0 + halfsel * 8) * OUT_DIM + n0 + l15;
        #pragma unroll
        for (int r = 0; r < 8; ++r)
            srow[(size_t)r * OUT_DIM] = acc[r];
    }
}

// ---------------------------------------------------------------------------
// Phase 2: out[dst[e]] += edge_weight[e] * support[src[e]]
//   one wave per edge; lane covers 4 columns (b128 gather + 4 f32 atomics)
//   support & out both fit in the 192MB L2 -> atomics resolve in L2
// ---------------------------------------------------------------------------
__global__ __launch_bounds__(256) void gcn_scatter(const float* __restrict__ S,
                                                   const float* __restrict__ ew,
                                                   const int* __restrict__ src,
                                                   const int* __restrict__ dst,
                                                   float* __restrict__ out) {
    int t    = blockIdx.x * 256 + threadIdx.x;
    int e    = t >> 5;
    int lane = t & 31;
    if (e >= N_EDGES) return;

    int   s = src[e];
    int   d = dst[e];
    float w = ew[e];

    v4f v = ((const v4f*)(S + (size_t)s * OUT_DIM))[lane];
    float* op = out + (size_t)d * OUT_DIM + lane * 4;

    unsafeAtomicAdd(op + 0, v.x * w);
    unsafeAtomicAdd(op + 1, v.y * w);
    unsafeAtomicAdd(op + 2, v.z * w);
    unsafeAtomicAdd(op + 3, v.w * w);
}

// ---------------------------------------------------------------------------
extern "C" void kernel_launch(void* const* d_in, const int* in_sizes, int n_in,
                              void* d_out, int out_size, void* d_ws, size_t ws_size,
                              hipStream_t stream) {
    const float* features    = (const float*)d_in[0];
    const float* weight      = (const float*)d_in[1];
    const float* bias        = (const float*)d_in[2];
    const float* edge_weight = (const float*)d_in[3];
    const int*   src         = (const int*)d_in[4];
    const int*   dst         = (const int*)d_in[5];
    float*       out         = (float*)d_out;
    float*       support     = (float*)d_ws;         // 100000*128*4 = 51.2 MB

    // out = broadcast(bias)
    gcn_init_out<<<N_NODES * OUT_DIM / 4 / 256, 256, 0, stream>>>(bias, out);

    // support = X @ W   (WMMA f32)
    gcn_gemm<<<2048, 256, 0, stream>>>(features, weight, support);

    // out[dst] += ew * support[src]
    gcn_scatter<<<(N_EDGES * 32) / 256, 256, 0, stream>>>(support, edge_weight,
                                                          src, dst, out);
}